// G2GCNModel_5858335391831
// MI455X (gfx1250) — compile-verified
//
#include <hip/hip_runtime.h>
#include <hip/hip_bf16.h>

typedef __attribute__((ext_vector_type(2))) float v2f;
typedef __attribute__((ext_vector_type(8))) float v8f;

#define CH 128   // hidden channels
#define C4 32    // float4 groups per row

// ---------------------------------------------------------------------------
// Y[N,CO] = X[N,K] @ W[K,CO] (+bias). One wave per 16x16 tile, fp32 WMMA.
// K, CO are compile-time so the 32-step K loop fully unrolls and every load
// becomes base-pointer + immediate offset (clauses of global_load feeding
// back-to-back v_wmma_f32_16x16x4_f32).
// blockDim = 128 (4 waves); wave w covers cols [(blockIdx.y*4+w)*16, +16).
// ---------------------------------------------------------------------------
template <int K, int CO>
__global__ void gemm_wmma(const float* __restrict__ X, const float* __restrict__ W,
                          const float* __restrict__ bias, float* __restrict__ Y,
                          int N) {
    const int wave = threadIdx.x >> 5;
    const int lane = threadIdx.x & 31;
    const int mBase   = blockIdx.x * 16;
    const int colBase = (blockIdx.y * 4 + wave) * 16;
    if (mBase >= N) return;                       // wave-uniform

    const int lh = lane >> 4;                     // 0: K pair {0,1}; 1: {2,3}
    const int ll = lane & 15;
    const int mLoad = (mBase + ll < N) ? (mBase + ll) : (N - 1);

    // Per-lane base pointers; all K-loop accesses are immediate offsets.
    const float* __restrict__ xbase = X + (size_t)mLoad * K + lh * 2;
    const float* __restrict__ wbase = W + (size_t)(lh * 2) * CO + colBase + ll;

    v8f c = {0.f, 0.f, 0.f, 0.f, 0.f, 0.f, 0.f, 0.f};
    #pragma unroll
    for (int kk = 0; kk < K; kk += 4) {
        v2f a = *reinterpret_cast<const v2f*>(xbase + kk);  // A: M=ll, K=kk+2lh{,+1}
        v2f b;
        b.x = wbase[kk * CO];                                // B row kk+2lh
        b.y = wbase[kk * CO + CO];                           // B row kk+2lh+1
        c = __builtin_amdgcn_wmma_f32_16x16x4_f32(
                /*neg_a=*/false, a, /*neg_b=*/false, b,
                /*c_mod=*/(short)0, c, /*reuse_a=*/false, /*reuse_b=*/false);
    }

    const float bv = bias ? bias[colBase + ll] : 0.0f;
    const int row0 = mBase + lh * 8;                         // C: VGPR r -> rows r, r+8
    #pragma unroll
    for (int r = 0; r < 8; ++r) {
        const int row = row0 + r;
        if (row < N) Y[(size_t)row * CO + colBase + ll] = c[r] + bv;
    }
}

// ---------------------------------------------------------------------------
__global__ void zero_f(float* __restrict__ p, int n) {
    int i = blockIdx.x * blockDim.x + threadIdx.x;
    if (i < n) p[i] = 0.0f;
}

__global__ void count_deg(const int* __restrict__ rows, float* __restrict__ deg, int E) {
    int e = blockIdx.x * blockDim.x + threadIdx.x;
    if (e < E) atomicAdd(&deg[rows[e]], 1.0f);
}

// agg[n][:] = bias[:]; tau_vals[n] = 0
__global__ void init_agg(float* __restrict__ agg, const float* __restrict__ bias,
                         float* __restrict__ tau_vals, int N) {
    int t = blockIdx.x * blockDim.x + threadIdx.x;           // float4 units
    if (t < N * C4) {
        float4 bb = reinterpret_cast<const float4*>(bias)[t & (C4 - 1)];
        reinterpret_cast<float4*>(agg)[t] = bb;
    }
    if (t < N) tau_vals[t] = 0.0f;
}

// agg[col[e]] += h[row[e]]  (4 channels / thread; f32 atomics resolve in L2:
// agg is 51.2 MB and stays resident in the 192 MB L2)
__global__ void scatter_add(const float* __restrict__ h, const int* __restrict__ rows,
                            const int* __restrict__ cols, float* __restrict__ agg, int E) {
    unsigned int t = blockIdx.x * blockDim.x + threadIdx.x;
    if (t >= (unsigned int)E * C4) return;
    int e  = (int)(t >> 5);
    int c4 = (int)(t & (C4 - 1));
    float4 v = reinterpret_cast<const float4*>(h + (size_t)rows[e] * CH)[c4];
    float* dst = agg + (size_t)cols[e] * CH + c4 * 4;
    atomicAdd(dst + 0, v.x);
    atomicAdd(dst + 1, v.y);
    atomicAdd(dst + 2, v.z);
    atomicAdd(dst + 3, v.w);
}

// One wave per edge: ||relu(agg[row]) - relu(agg[col])||^2 (ReLU fused into
// the gather), shfl_xor tree reduce, single atomic per edge.
__global__ void edge_tau(const float* __restrict__ agg, const int* __restrict__ rows,
                         const int* __restrict__ cols, float* __restrict__ tau_vals, int E) {
    int e = blockIdx.x * (blockDim.x >> 5) + (threadIdx.x >> 5);
    if (e >= E) return;                                      // wave-uniform
    int lane = threadIdx.x & 31;
    int r = rows[e], c = cols[e];
    float4 a = reinterpret_cast<const float4*>(agg + (size_t)r * CH)[lane];
    float4 b = reinterpret_cast<const float4*>(agg + (size_t)c * CH)[lane];
    a.x = fmaxf(a.x, 0.f); a.y = fmaxf(a.y, 0.f); a.z = fmaxf(a.z, 0.f); a.w = fmaxf(a.w, 0.f);
    b.x = fmaxf(b.x, 0.f); b.y = fmaxf(b.y, 0.f); b.z = fmaxf(b.z, 0.f); b.w = fmaxf(b.w, 0.f);
    float dx = a.x - b.x, dy = a.y - b.y, dz = a.z - b.z, dw = a.w - b.w;
    float s = dx * dx + dy * dy + dz * dz + dw * dw;
    #pragma unroll
    for (int off = 16; off > 0; off >>= 1) s += __shfl_xor(s, off, 32);
    if (lane == 0) atomicAdd(&tau_vals[r], s);
}

// x = x + tau*(relu(agg) - x),  tau = tanh(tau_vals/(deg+1e-10))
__global__ void update_x(float* __restrict__ x, const float* __restrict__ agg,
                         const float* __restrict__ tau_vals, const float* __restrict__ deg,
                         int N) {
    int t = blockIdx.x * blockDim.x + threadIdx.x;           // float4 units
    if (t >= N * C4) return;
    int n = t >> 5;
    float tau = tanhf(tau_vals[n] / (deg[n] + 1e-10f));
    float4 xo = reinterpret_cast<float4*>(x)[t];
    float4 xn = reinterpret_cast<const float4*>(agg)[t];
    xo.x += tau * (fmaxf(xn.x, 0.f) - xo.x);
    xo.y += tau * (fmaxf(xn.y, 0.f) - xo.y);
    xo.z += tau * (fmaxf(xn.z, 0.f) - xo.z);
    xo.w += tau * (fmaxf(xn.w, 0.f) - xo.w);
    reinterpret_cast<float4*>(x)[t] = xo;
}

// ---------------------------------------------------------------------------
extern "C" void kernel_launch(void* const* d_in, const int* in_sizes, int n_in,
                              void* d_out, int out_size, void* d_ws, size_t ws_size,
                              hipStream_t stream) {
    const float* x_in    = (const float*)d_in[0];
    const int*   eidx    = (const int*)  d_in[1];   // [2,E] row-major: rows then cols
    const float* W_in    = (const float*)d_in[2];
    const float* W_convs = (const float*)d_in[3];
    const float* b_convs = (const float*)d_in[4];
    const float* W_fc    = (const float*)d_in[5];
    const float* b_fc    = (const float*)d_in[6];

    const int N = in_sizes[0] / CH;     // 100000
    const int E = in_sizes[1] / 2;      // 1600000
    const int* rows = eidx;
    const int* cols = eidx + E;

    float* ws   = (float*)d_ws;
    float* xcur = ws;                                  // N*128
    float* h    = ws + (size_t)N * CH;                 // N*128
    float* agg  = ws + (size_t)N * CH * 2;             // N*128 (pre-ReLU x_new)
    float* tauv = ws + (size_t)N * CH * 3;             // N
    float* deg  = tauv + N;                            // N

    const int mTiles = (N + 15) / 16;
    const dim3 gblk(128);                              // 4 waves

    // x = x @ W_in
    gemm_wmma<CH, CH><<<dim3(mTiles, 2), gblk, 0, stream>>>(x_in, W_in, nullptr, xcur, N);

    // degree (constant across layers)
    zero_f<<<(N + 255) / 256, 256, 0, stream>>>(deg, N);
    count_deg<<<(E + 255) / 256, 256, 0, stream>>>(rows, deg, E);

    const unsigned int scatterBlocks = ((unsigned int)E * C4 + 255u) / 256u;
    const int n4Blocks = (N * C4 + 255) / 256;
    const int edgeWaveBlocks = (E + 7) / 8;            // 8 waves / 256-thread block

    for (int l = 0; l < 3; ++l) {
        // h = x @ W_convs[l]
        gemm_wmma<CH, CH><<<dim3(mTiles, 2), gblk, 0, stream>>>(
            xcur, W_convs + (size_t)l * CH * CH, nullptr, h, N);
        // agg = b_conv[l]; tau_vals = 0
        init_agg<<<n4Blocks, 256, 0, stream>>>(agg, b_convs + (size_t)l * CH, tauv, N);
        // agg[col] += h[row]
        scatter_add<<<scatterBlocks, 256, 0, stream>>>(h, rows, cols, agg, E);
        // tau_vals[row] += ||relu(agg[row]) - relu(agg[col])||^2
        edge_tau<<<edgeWaveBlocks, 256, 0, stream>>>(agg, rows, cols, tauv, E);
        // x = (1-tau)*x + tau*relu(agg)
        update_x<<<n4Blocks, 256, 0, stream>>>(xcur, agg, tauv, deg, N);
    }

    // out = x @ W_fc + b_fc
    gemm_wmma<CH, 64><<<dim3(mTiles, 1), gblk, 0, stream>>>(
        xcur, W_fc, b_fc, (float*)d_out, N);
}